// SLayer_6545530159284
// MI455X (gfx1250) — compile-verified
//
#include <hip/hip_runtime.h>

typedef float v2f __attribute__((ext_vector_type(2)));
typedef float v8f __attribute__((ext_vector_type(8)));

#define B_    128
#define P_    4096
#define N_    64
#define SPLIT 8
#define PPB   (P_ / SPLIT)   // 512 points per block

// dist[p,n] = base[n] + k0[n]*x0 + k1[n]*x1 + k2[n]*x0^2 + k3[n]*x1^2
// All coefficients pre-scaled by -log2(e) so exp(-dist) == exp2(q).
// base[n] is folded into the WMMA C operand, so q comes straight out of the
// matrix unit: q = A(features) x B(coeffs) + C(base).
__global__ __launch_bounds__(256)
void slayer_main(const float* __restrict__ batch,      // [B,P,2]
                 const float* __restrict__ mask,       // [B,P]
                 const float* __restrict__ centers,    // [N,2]
                 const float* __restrict__ sharp,      // [N,2]
                 float* __restrict__ partial)          // [B,SPLIT,N]
{
    const int s    = blockIdx.x;
    const int b    = blockIdx.y;
    const int tid  = threadIdx.x;
    const int lane = tid & 31;
    const int wave = tid >> 5;
    const int col  = lane & 15;
    const bool hi  = lane >= 16;

    const float NEG_LOG2E = -1.44269504088896340736f;

    // ---- Hoisted: B-matrix operands (4 n-tiles) + base folded into C ----
    // B layout (4x16 f32): lane L holds column n = t*16 + (L%16);
    // VGPR0 = K(L<16?0:2), VGPR1 = K(L<16?1:3).
    v2f bmat[4];
    v8f cbase[4];
#pragma unroll
    for (int t = 0; t < 4; ++t) {
        const int n = t * 16 + col;
        const float c0 = centers[n * 2 + 0], c1 = centers[n * 2 + 1];
        const float s0 = sharp[n * 2 + 0],   s1 = sharp[n * 2 + 1];
        const float s0sq = s0 * s0, s1sq = s1 * s1;
        const float k0 = NEG_LOG2E * (-2.0f * s0sq * c0);
        const float k1 = NEG_LOG2E * (-2.0f * s1sq * c1);
        const float k2 = NEG_LOG2E * s0sq;
        const float k3 = NEG_LOG2E * s1sq;
        bmat[t][0] = hi ? k2 : k0;
        bmat[t][1] = hi ? k3 : k1;
        const float base = NEG_LOG2E * (s0sq * c0 * c0 + s1sq * c1 * c1);
#pragma unroll
        for (int r = 0; r < 8; ++r) cbase[t][r] = base;  // same col -> same base
    }

    float acc[4] = {0.f, 0.f, 0.f, 0.f};

    const float2* bat = (const float2*)batch + (size_t)b * P_;
    const float*  msk = mask + (size_t)b * P_;
    const int p_blk = s * PPB;

#pragma unroll 1
    for (int it = 0; it < PPB / 128; ++it) {
        const int p0 = p_blk + it * 128 + wave * 16;

        // A operand (16x4 f32): lane L holds point M = L%16;
        // low lanes carry (x0, x1) [K0,K1], high lanes carry (x0^2, x1^2) [K2,K3].
        const float2 xv = bat[p0 + col];
        v2f a;
        a[0] = hi ? xv.x * xv.x : xv.x;
        a[1] = hi ? xv.y * xv.y : xv.y;

        // Mask values for this half's 8 rows (M = r or r+8), contiguous in memory.
        const float4 m4a = *(const float4*)(msk + p0 + (hi ? 8 : 0));
        const float4 m4b = *(const float4*)(msk + p0 + (hi ? 8 : 0) + 4);
        const float m[8] = {m4a.x, m4a.y, m4a.z, m4a.w,
                            m4b.x, m4b.y, m4b.z, m4b.w};

#pragma unroll
        for (int t = 0; t < 4; ++t) {
            // q = features x coeffs + base, straight from the matrix unit.
            const v8f q = __builtin_amdgcn_wmma_f32_16x16x4_f32(
                    /*neg_a=*/false, a, /*neg_b=*/false, bmat[t],
                    /*c_mod=*/(short)0, cbase[t],
                    /*reuse_a=*/false, /*reuse_b=*/false);

            // 8 independent exps first (keeps the TRANS pipe busy, no
            // exp->fma hazard nops), then a 2-accumulator FMA reduction.
            float e[8];
#pragma unroll
            for (int r = 0; r < 8; ++r) e[r] = __builtin_amdgcn_exp2f(q[r]);

            float s0 = 0.f, s1 = 0.f;
#pragma unroll
            for (int r = 0; r < 8; r += 2) {
                s0 = fmaf(e[r],     m[r],     s0);
                s1 = fmaf(e[r + 1], m[r + 1], s1);
            }
            acc[t] += s0 + s1;
        }
    }

    // ---- Deterministic cross-lane / cross-wave reduction via LDS ----
    __shared__ float lds[8 * 4 * 32];
#pragma unroll
    for (int t = 0; t < 4; ++t)
        lds[wave * 128 + t * 32 + lane] = acc[t];
    __syncthreads();

    if (tid < 64) {
        const int t = tid >> 4, cc = tid & 15;
        float sum = 0.f;
#pragma unroll
        for (int w = 0; w < 8; ++w)
            sum += lds[w * 128 + t * 32 + cc] + lds[w * 128 + t * 32 + cc + 16];
        partial[((size_t)b * SPLIT + s) * N_ + tid] = sum;
    }
}

__global__ __launch_bounds__(256)
void slayer_reduce(const float* __restrict__ partial, float* __restrict__ out)
{
    const int i = blockIdx.x * blockDim.x + threadIdx.x;   // i = b*64 + n
    if (i < B_ * N_) {
        const int b = i >> 6, n = i & 63;
        float s = 0.f;
#pragma unroll
        for (int k = 0; k < SPLIT; ++k)
            s += partial[((size_t)b * SPLIT + k) * N_ + n];
        out[i] = s;
    }
}

extern "C" void kernel_launch(void* const* d_in, const int* in_sizes, int n_in,
                              void* d_out, int out_size, void* d_ws, size_t ws_size,
                              hipStream_t stream)
{
    const float* batch   = (const float*)d_in[0];   // [128,4096,2]
    const float* mask    = (const float*)d_in[1];   // [128,4096]
    const float* centers = (const float*)d_in[2];   // [64,2]
    const float* sharp   = (const float*)d_in[3];   // [64,2]
    float* out     = (float*)d_out;                 // [128,64]
    float* partial = (float*)d_ws;                  // [128,8,64] = 256 KB

    dim3 grid(SPLIT, B_);
    slayer_main<<<grid, 256, 0, stream>>>(batch, mask, centers, sharp, partial);
    slayer_reduce<<<(B_ * N_ + 255) / 256, 256, 0, stream>>>(partial, out);
}